// CliffordTransformLayer_38190849196332
// MI455X (gfx1250) — compile-verified
//
#include <hip/hip_runtime.h>
#include <utility>
#include <cstddef>

typedef __attribute__((ext_vector_type(2))) float v2f;
typedef __attribute__((ext_vector_type(4))) float f4v;
typedef __attribute__((ext_vector_type(8))) float v8f;

// ---------------- compile-time Cl(4,1) blade algebra ----------------
namespace ga {

constexpr int pcnt(int x){ int c=0; while(x){ c += x & 1; x >>= 1; } return c; }

// geometric product sign of basis blades a*b; metric (+,+,+,+,-), e5 = bit4
constexpr int gp_sign(int a, int b){
  int s = 0; int t = a >> 1;
  while (t){ s += pcnt(t & b); t >>= 1; }
  int sign = (s & 1) ? -1 : 1;
  if (a & b & 16) sign = -sign;   // only e5 squares to -1
  return sign;
}
constexpr int rev_sign(int m){ int g = pcnt(m); return ((g*(g-1)/2) & 1) ? -1 : 1; }

struct Core {
  int evenb[16]; int oddb[16]; int oddIdx[32];
  // stage 1: 80 pairs m=(i,j): even_i * vec_j -> odd k, sign s
  int p1i[80], p1j[80], p1k[80], p1s[80];
  // stage 2: 80 grade-1 pairs (k,i), split by k<8 (h=0) / k>=8 (h=1): 40 each
  int qk[2][40], qi[2][40], ql[2][40], qs[2][40];
};

constexpr Core make_core(){
  Core C{};
  int ne = 0, no = 0;
  for (int m = 0; m < 32; m++){
    if (pcnt(m) % 2 == 0) C.evenb[ne++] = m;
    else { C.oddb[no] = m; C.oddIdx[m] = no; no++; }
  }
  for (int i = 0; i < 16; i++)
    for (int j = 0; j < 5; j++){
      int m = i*5 + j;
      int e = C.evenb[i], vb = 1 << j;
      C.p1i[m] = i; C.p1j[m] = j;
      C.p1k[m] = C.oddIdx[e ^ vb];
      C.p1s[m] = gp_sign(e, vb);
    }
  int cnt0 = 0, cnt1 = 0;
  for (int k = 0; k < 16; k++){
    int o = C.oddb[k];
    for (int i = 0; i < 16; i++){
      int e = C.evenb[i];
      int blade = o ^ e;
      if (pcnt(blade) != 1) continue;         // keep grade-1 only
      int h = (k >= 8) ? 1 : 0;
      int idx = h ? cnt1++ : cnt0++;
      C.qk[h][idx] = k; C.qi[h][idx] = i;
      int l = 0; while (!((blade >> l) & 1)) l++;
      C.ql[h][idx] = l;
      C.qs[h][idx] = gp_sign(o, e) * rev_sign(e);
    }
  }
  return C;
}

constexpr Core CORE = make_core();

// Constant A-operands (sign matrices) in V_WMMA_F32_16X16X4_F32 A-layout:
// lane L: M = L&15, half h = L>>4; VGPR r holds K = 4c + r + 2h.
// Layout transposed to [lane][chunk][2]: each lane's 40 float2 operands are a
// contiguous 320B run -> one SGPR base + lane*320, immediate per-chunk offsets.
struct ATabs { float a[32][40][2]; };   // chunk 0..19 = stage1, 20..39 = stage2

constexpr ATabs make_atabs(){
  ATabs T{};
  for (int lane = 0; lane < 32; lane++){
    int h = lane >> 4, M = lane & 15;
    for (int c = 0; c < 20; c++)
      for (int r = 0; r < 2; r++){
        int m = 4*c + r + 2*h;                       // stage-1 K slot -> pair m
        T.a[lane][c][r] = (M == CORE.p1k[m]) ? (float)CORE.p1s[m] : 0.0f;
        int idx = 2*c + r;                           // stage-2: half h uses list h
        T.a[lane][20 + c][r] = (M == CORE.ql[h][idx]) ? (float)CORE.qs[h][idx] : 0.0f;
      }
  }
  return T;
}

} // namespace ga

__constant__ ga::ATabs cT = ga::make_atabs();

// ---------------- WMMA chunk helpers ----------------
template<int C>
__device__ __forceinline__ v8f s1_chunk(v8f acc, const float (&vv)[16],
                                        const float (&pv)[5], bool hi,
                                        const char* __restrict__ tl){
  // B operand: W^T rows K = 4C + {0,1} (low half) / {2,3} (high half); N = lane&15
  constexpr int mL0 = 4*C + 0, mL1 = 4*C + 1, mH0 = 4*C + 2, mH1 = 4*C + 3;
  constexpr int iL0 = ga::CORE.p1i[mL0], jL0 = ga::CORE.p1j[mL0];
  constexpr int iL1 = ga::CORE.p1i[mL1], jL1 = ga::CORE.p1j[mL1];
  constexpr int iH0 = ga::CORE.p1i[mH0], jH0 = ga::CORE.p1j[mH0];
  constexpr int iH1 = ga::CORE.p1i[mH1], jH1 = ga::CORE.p1j[mH1];
  float xl = vv[iL0] * pv[jL0], yl = vv[iL1] * pv[jL1];
  float xh = vv[iH0] * pv[jH0], yh = vv[iH1] * pv[jH1];
  v2f b; b.x = hi ? xh : xl; b.y = hi ? yh : yl;
  v2f a = *(const v2f*)(tl + C * 8);                // single base, imm offset
  return __builtin_amdgcn_wmma_f32_16x16x4_f32(false, a, false, b, (short)0, acc,
                                               false, false);
}

template<int C>
__device__ __forceinline__ v8f s2_chunk(v8f acc, const v8f& t, const float (&vv)[16],
                                        bool hi, const char* __restrict__ tl){
  // lane<16 holds tmp[n, k=0..7] in t[0..7]; lane>=16 holds tmp[n, k=8..15]
  constexpr int kL0 = ga::CORE.qk[0][2*C+0], iL0 = ga::CORE.qi[0][2*C+0];
  constexpr int kL1 = ga::CORE.qk[0][2*C+1], iL1 = ga::CORE.qi[0][2*C+1];
  constexpr int kH0 = ga::CORE.qk[1][2*C+0], iH0 = ga::CORE.qi[1][2*C+0];
  constexpr int kH1 = ga::CORE.qk[1][2*C+1], iH1 = ga::CORE.qi[1][2*C+1];
  float xl = t[kL0]     * vv[iL0], yl = t[kL1]     * vv[iL1];
  float xh = t[kH0 - 8] * vv[iH0], yh = t[kH1 - 8] * vv[iH1];
  v2f b; b.x = hi ? xh : xl; b.y = hi ? yh : yl;
  v2f a = *(const v2f*)(tl + (20 + C) * 8);         // stage-2 half of the run
  return __builtin_amdgcn_wmma_f32_16x16x4_f32(false, a, false, b, (short)0, acc,
                                               false, false);
}

template<size_t... Cs>
__device__ __forceinline__ v8f run_s1(std::index_sequence<Cs...>, const float (&vv)[16],
                                      const float (&pv)[5], bool hi, const char* tl){
  v8f acc = (v8f)0.0f;
  ((acc = s1_chunk<(int)Cs>(acc, vv, pv, hi, tl)), ...);
  return acc;
}
template<size_t... Cs>
__device__ __forceinline__ v8f run_s2(std::index_sequence<Cs...>, const v8f& t,
                                      const float (&vv)[16], bool hi, const char* tl){
  v8f acc = (v8f)0.0f;
  ((acc = s2_chunk<(int)Cs>(acc, t, vv, hi, tl)), ...);
  return acc;
}

// ---------------- kernel: one wave = one 16-element tile ----------------
__global__ __launch_bounds__(256) void clifford_sandwich_wmma(
    const float* __restrict__ versor, const float* __restrict__ point,
    float* __restrict__ out, int ntiles, int nElem){
  int tid  = blockIdx.x * blockDim.x + threadIdx.x;
  int wave = tid >> 5;
  if (wave >= ntiles) return;            // wave-uniform exit; EXEC full inside
  int lane = threadIdx.x & 31;
  bool hi  = lane >= 16;
  int n  = wave * 16 + (lane & 15);
  int nc = (n < nElem) ? n : (nElem - 1);  // clamp tail loads (dup reads harmless)

  // per-lane base into the constant sign-operand table (320 B per lane)
  const char* tl = (const char*)&cT + (size_t)lane * 320;

  float vv[16];
  const f4v* vp = (const f4v*)(versor + (size_t)nc * 16);
  f4v v0 = vp[0], v1 = vp[1], v2 = vp[2], v3 = vp[3];
  vv[0]=v0.x; vv[1]=v0.y; vv[2]=v0.z; vv[3]=v0.w;
  vv[4]=v1.x; vv[5]=v1.y; vv[6]=v1.z; vv[7]=v1.w;
  vv[8]=v2.x; vv[9]=v2.y; vv[10]=v2.z; vv[11]=v2.w;
  vv[12]=v3.x; vv[13]=v3.y; vv[14]=v3.z; vv[15]=v3.w;

  float pv[5];
  const float* pp = point + (size_t)nc * 5;
  pv[0]=pp[0]; pv[1]=pp[1]; pv[2]=pp[2]; pv[3]=pp[3]; pv[4]=pp[4];

  // TMP^T = S1^T x W^T  (20 chained v_wmma_f32_16x16x4_f32)
  v8f t = run_s1(std::make_index_sequence<20>{}, vv, pv, hi, tl);
  // OUT^T = S2^T x U^T  (20 more)
  v8f o = run_s2(std::make_index_sequence<20>{}, t, vv, hi, tl);

  // lane n (<16) holds out[n, l] in o[l], l = 0..4
  if (!hi && n < nElem){
    float* op = out + (size_t)n * 5;
    op[0]=o[0]; op[1]=o[1]; op[2]=o[2]; op[3]=o[3]; op[4]=o[4];
  }
}

extern "C" void kernel_launch(void* const* d_in, const int* in_sizes, int n_in,
                              void* d_out, int out_size, void* d_ws, size_t ws_size,
                              hipStream_t stream) {
  const float* versor = (const float*)d_in[0];
  const float* point  = (const float*)d_in[1];
  float* out = (float*)d_out;
  int N = in_sizes[1] / 5;                   // number of elements
  int ntiles = (N + 15) / 16;
  int threads = 256;                          // 8 wave32 per block
  int wpb = threads / 32;
  int blocks = (ntiles + wpb - 1) / wpb;
  clifford_sandwich_wmma<<<blocks, threads, 0, stream>>>(versor, point, out, ntiles, N);
}